// PositionalAPP_6416681140617
// MI455X (gfx1250) — compile-verified
//
#include <hip/hip_runtime.h>
#include <hip/hip_bf16.h>

typedef __attribute__((ext_vector_type(2))) float v2f;
typedef __attribute__((ext_vector_type(8))) float v8f;

#define BATCH 32
#define CH    512
#define HW    4096          // 64*64
#define BN_EPS 1e-5f

// ---------------------------------------------------------------------------
// Kernel 1: per-channel mean of bilinear-upsampled (7x7 -> 64x64) pos_embed.
// Half-pixel centers, edge clamp (== jax.image.resize / torch align_corners=F).
// 512 blocks x 64 threads; thread t integrates output row t. Runs first so
// gap_kernel can fold posm into pooled (keeps the WMMA loop lean).
// ---------------------------------------------------------------------------
__global__ void __launch_bounds__(64) posmean_kernel(const float* __restrict__ pe,
                                                     float* __restrict__ posm) {
    const int c = blockIdx.x;
    __shared__ float tile[49];
    if (threadIdx.x < 49) tile[threadIdx.x] = pe[c * 49 + threadIdx.x];
    __syncthreads();

    const float scale = 7.0f / 64.0f;
    const int y = threadIdx.x;                          // 0..63
    float sy = (y + 0.5f) * scale - 0.5f;
    float fy = floorf(sy);
    int   y0 = (int)fy;
    float wy = sy - fy;
    int y0c = min(6, max(0, y0));
    int y1c = min(6, max(0, y0 + 1));

    float s = 0.f;
    for (int xc = 0; xc < 64; ++xc) {
        float sx = (xc + 0.5f) * scale - 0.5f;
        float fx = floorf(sx);
        int   x0 = (int)fx;
        float wx = sx - fx;
        int x0c = min(6, max(0, x0));
        int x1c = min(6, max(0, x0 + 1));
        float top = (1.f - wx) * tile[y0c * 7 + x0c] + wx * tile[y0c * 7 + x1c];
        float bot = (1.f - wx) * tile[y1c * 7 + x0c] + wx * tile[y1c * 7 + x1c];
        s += (1.f - wy) * top + wy * bot;
    }
#pragma unroll
    for (int off = 16; off > 0; off >>= 1)
        s += __shfl_down(s, off, 32);
    __shared__ float lds2[2];
    if ((threadIdx.x & 31) == 0) lds2[threadIdx.x >> 5] = s;
    __syncthreads();
    if (threadIdx.x == 0) posm[c] = (lds2[0] + lds2[1]) * (1.0f / (float)HW);
}

// ---------------------------------------------------------------------------
// Kernel 2: global average pool over each (b,c) 64x64 plane, + posm[c].
// 16384 blocks x 256 threads; 4x float4 (b128) loads per thread, coalesced.
// This is the HBM-bound part: 268 MB read ~= 11.5 us at 23.3 TB/s.
// ---------------------------------------------------------------------------
__global__ void __launch_bounds__(256) gap_kernel(const float* __restrict__ x,
                                                  const float* __restrict__ posm,
                                                  float* __restrict__ pooled) {
    const int plane = blockIdx.x;                       // b*CH + c
    const float4* p = (const float4*)(x + (size_t)plane * HW);
    float s = 0.f;
#pragma unroll
    for (int j = 0; j < 4; ++j) {
        float4 v = p[j * 256 + threadIdx.x];
        s += (v.x + v.y) + (v.z + v.w);
    }
    // wave32 reduce
#pragma unroll
    for (int off = 16; off > 0; off >>= 1)
        s += __shfl_down(s, off, 32);
    __shared__ float lds[8];
    if ((threadIdx.x & 31) == 0) lds[threadIdx.x >> 5] = s;
    __syncthreads();
    if (threadIdx.x == 0) {
        float t = 0.f;
#pragma unroll
        for (int w = 0; w < 8; ++w) t += lds[w];
        pooled[plane] = t * (1.0f / (float)HW) + posm[plane & (CH - 1)];
    }
}

// ---------------------------------------------------------------------------
// Kernel 3: y = relu(BN(pooled @ conv_w^T + conv_b)) via V_WMMA_F32_16X16X4_F32.
// One wave per 16x16 output tile; K=512 -> 128 WMMAs per wave.
// A layout (16x4 f32): lane l(0-15)=M, VGPR0/1 = K = 2*half + {0,1}.
// B layout (4x16 f32): lane l = N,      VGPR0/1 = K = 2*half + {0,1}.
//   B[k][n] = conv_w[n][k] -> contiguous (8B-aligned) float2 from row-major W.
// C/D layout: VGPR r -> M = r + 8*half, lane -> N.
// Loads are batched 16 k-steps at a time (32 outstanding b64 loads, one wait,
// then 16 back-to-back WMMAs) so exposed latency is ~8 round-trips total.
// ---------------------------------------------------------------------------
__global__ void __launch_bounds__(32) gemm_bn_relu_kernel(
    const float* __restrict__ pooled, const float* __restrict__ W,
    const float* __restrict__ bias,   const float* __restrict__ gamma,
    const float* __restrict__ beta,   const float* __restrict__ mean,
    const float* __restrict__ var,    float* __restrict__ out) {
    const int tile  = blockIdx.x;                 // 0..63
    const int tileM = (tile & 1) * 16;            // 2 M-tiles (B=32)
    const int tileN = (tile >> 1) * 16;           // 32 N-tiles (O=512)
    const int lane  = threadIdx.x;
    const int half  = lane >> 4;
    const int l     = lane & 15;

    const float* Arow = pooled + (size_t)(tileM + l) * CH + 2 * half;
    const float* Brow = W      + (size_t)(tileN + l) * CH + 2 * half;

    v8f acc = {};
    for (int k0 = 0; k0 < CH; k0 += 64) {         // 8 chunks of 16 k-steps
        v2f a[16], b[16];
#pragma unroll
        for (int i = 0; i < 16; ++i) {
            a[i] = *(const v2f*)(Arow + k0 + 4 * i);
            b[i] = *(const v2f*)(Brow + k0 + 4 * i);
        }
#pragma unroll
        for (int i = 0; i < 16; ++i) {
            acc = __builtin_amdgcn_wmma_f32_16x16x4_f32(
                /*neg_a=*/false, a[i], /*neg_b=*/false, b[i],
                /*c_mod=*/(short)0, acc, /*reuse_a=*/false, /*reuse_b=*/false);
        }
    }

    const int n  = tileN + l;
    const float sc = gamma[n] * rsqrtf(var[n] + BN_EPS);
    const float sh = beta[n] - mean[n] * sc;
    const float bi = bias[n];
#pragma unroll
    for (int r = 0; r < 8; ++r) {
        const int m = tileM + r + 8 * half;
        float v = (acc[r] + bi) * sc + sh;
        out[m * CH + n] = fmaxf(v, 0.f);
    }
}

extern "C" void kernel_launch(void* const* d_in, const int* in_sizes, int n_in,
                              void* d_out, int out_size, void* d_ws, size_t ws_size,
                              hipStream_t stream) {
    const float* x   = (const float*)d_in[0];   // (32,512,64,64)
    const float* pe  = (const float*)d_in[1];   // (1,512,7,7)
    const float* W   = (const float*)d_in[2];   // (512,512) row-major (O,C)
    const float* b   = (const float*)d_in[3];   // (512,)
    const float* g   = (const float*)d_in[4];   // bn_gamma
    const float* bt  = (const float*)d_in[5];   // bn_beta
    const float* mn  = (const float*)d_in[6];   // bn_mean
    const float* vr  = (const float*)d_in[7];   // bn_var
    float* out = (float*)d_out;                 // (32,512,1,1) flat

    float* pooled = (float*)d_ws;               // 32*512 floats
    float* posm   = pooled + BATCH * CH;        // 512 floats

    posmean_kernel<<<CH, 64, 0, stream>>>(pe, posm);
    gap_kernel<<<BATCH * CH, 256, 0, stream>>>(x, posm, pooled);
    gemm_bn_relu_kernel<<<(BATCH / 16) * (CH / 16), 32, 0, stream>>>(
        pooled, W, b, g, bt, mn, vr, out);
}